// Conv2d_27436251087256
// MI455X (gfx1250) — compile-verified
//
#include <hip/hip_runtime.h>

// ---------------------------------------------------------------------------
// SubMConv2d on fully-active grid == dense 3x3 SAME conv, stride 1, no bias.
//   x   : (B=8, H=1024, W=1024, C=16) fp32  (raw NHWC view of the flat buffer)
//   W   : (3, 3, Cin=16, Cout=16)      fp32 (HWIO)
//   out : (B=8, Cout=16, H=1024, W=1024) fp32 (NCHW)
//
// Implicit GEMM with V_WMMA_F32_16X16X4_F32 (fp32 A/B, fp32 accum — full
// precision vs the fp32 reference). One wave computes a 16-pixel x 16-cout
// tile; K = 9 taps * 16 cin = 36 WMMAs of K=4, fully utilized.
// ---------------------------------------------------------------------------

typedef __attribute__((ext_vector_type(2))) float v2f;
typedef __attribute__((ext_vector_type(8))) float v8f;

#define HS   1024
#define WS   1024
#define CIN  16
#define COUT 16

#define SLAB_PX 130   // 128-pixel block segment + 2 halo columns
#define PXS     20    // padded floats per pixel in LDS (bank-conflict-free A loads)

__global__ __launch_bounds__(256)
void subm_conv3x3_wmma_f32(const float* __restrict__ x,
                           const float* __restrict__ wgt,
                           float* __restrict__ out)
{
    // 3 rows x 130 px x 20 floats = 7800 floats (31.2 KB)
    __shared__ float slab[3 * SLAB_PX * PXS];
    // B fragments pre-swizzled: 9 taps x 4 kchunks x 2 halves x 16 co x 2 = 2304 floats (9.2 KB)
    __shared__ float wl[9 * 4 * 2 * 16 * 2];

    const int tid  = threadIdx.x;
    const int lane = tid & 31;
    const int wave = tid >> 5;

    const int blk   = blockIdx.x;
    const int seg   = blk & 7;            // which 128-px segment of the row
    const int h     = (blk >> 3) & 1023;  // output row
    const int b     = blk >> 13;          // batch
    const int w0blk = seg * 128;          // first output column of this block

    // ---- Stage weights, pre-swizzled into B-fragment lane order -------------
    // pair p: n = p&15 (co), half = (p>>4)&1 (K hi/lo lanes), q = (p>>5)&3, tap = p>>7
    // lane L of B-frag reads pair { W[tap][q*4 + (L>>4)*2][n], W[tap][...+1][n] }
    for (int p = tid; p < 9 * 4 * 2 * 16; p += 256) {
        const int n    = p & 15;
        const int half = (p >> 4) & 1;
        const int q    = (p >> 5) & 3;
        const int tap  = p >> 7;
        const int k    = q * 4 + half * 2;
        wl[p * 2 + 0] = wgt[(tap * CIN + k    ) * COUT + n];
        wl[p * 2 + 1] = wgt[(tap * CIN + k + 1) * COUT + n];
    }

    // ---- Stage input slab: rows h-1..h+1, cols w0blk-1 .. w0blk+128 ---------
    // zero-fill SAME padding at image borders. float4 granularity: 3*130*4 = 1560.
    for (int i = tid; i < 3 * SLAB_PX * 4; i += 256) {
        const int c4  = i & 3;
        const int px  = (i >> 2) % SLAB_PX;
        const int row = (i >> 2) / SLAB_PX;
        const int gw  = w0blk - 1 + px;
        const int gh  = h - 1 + row;
        float4 v = make_float4(0.f, 0.f, 0.f, 0.f);
        if ((unsigned)gw < (unsigned)WS && (unsigned)gh < (unsigned)HS) {
            v = *reinterpret_cast<const float4*>(
                    x + (((long)b * HS + gh) * WS + gw) * CIN + c4 * 4);
        }
        *reinterpret_cast<float4*>(&slab[(row * SLAB_PX + px) * PXS + c4 * 4]) = v;
    }

    __syncthreads();

    // ---- Compute: this wave's 16-px x 16-co tile ----------------------------
    const int m16 = wave * 16;       // tile's pixel offset inside the block
    const int mlo = lane & 15;       // A: M index / B: N index
    const int khi = lane >> 4;       // K lane-half (0 or 1)

    v8f acc = {};                    // fp32 accumulator (zero)

#pragma unroll
    for (int kh = 0; kh < 3; ++kh) {
        const float* srow = &slab[kh * SLAB_PX * PXS];
#pragma unroll
        for (int kw = 0; kw < 3; ++kw) {
            const int tap = kh * 3 + kw;
            // A: slab px = m16 + mlo + kw  (the -1 halo is baked into the slab)
            const float* apx = srow + (m16 + mlo + kw) * PXS + khi * 2;
            // B: swizzled pair base for (tap, q=0, half=khi, n=mlo)
            const float* bpx = &wl[((tap * 8 + khi) * 16 + mlo) * 2];
#pragma unroll
            for (int q = 0; q < 4; ++q) {
                v2f a = *reinterpret_cast<const v2f*>(apx + q * 4);        // ds_load_b64
                v2f bm = *reinterpret_cast<const v2f*>(bpx + q * 2 * 32);  // ds_load_b64
                // 8 args: (neg_a, A, neg_b, B, c_mod, C, reuse_a, reuse_b)
                acc = __builtin_amdgcn_wmma_f32_16x16x4_f32(
                        false, a, false, bm, (short)0, acc, false, false);
            }
        }
    }

    // ---- Store: lane holds co = mlo, pixels m = khi*8 + r (r = 0..7) --------
    // acc[r] -> output column w0blk + m16 + khi*8 + r : two aligned b128 stores.
    const int   co   = mlo;
    const int   wout = w0blk + m16 + khi * 8;
    float* op = out + (((long)b * COUT + co) * HS + h) * WS + wout;
    *reinterpret_cast<float4*>(op)     = make_float4(acc[0], acc[1], acc[2], acc[3]);
    *reinterpret_cast<float4*>(op + 4) = make_float4(acc[4], acc[5], acc[6], acc[7]);
}

extern "C" void kernel_launch(void* const* d_in, const int* in_sizes, int n_in,
                              void* d_out, int out_size, void* d_ws, size_t ws_size,
                              hipStream_t stream)
{
    const float* x   = (const float*)d_in[0];  // (8,16,1024,1024) viewed flat as NHWC
    const float* wgt = (const float*)d_in[1];  // (3,3,16,16) HWIO
    float*       out = (float*)d_out;          // (8,16,1024,1024) NCHW

    // grid: 8 segments/row * 1024 rows * 8 batches = 65536 blocks of 8 waves
    dim3 grid(8 * 1024 * 8);
    subm_conv3x3_wmma_f32<<<grid, 256, 0, stream>>>(x, wgt, out);
}